// GNNEncoder_11416023073365
// MI455X (gfx1250) — compile-verified
//
#include <hip/hip_runtime.h>
#include <hip/hip_bf16.h>

typedef float v2f __attribute__((ext_vector_type(2)));
typedef float v8f __attribute__((ext_vector_type(8)));

#define HDIM 64
#define GDIM 256

// ---------------------------------------------------------------------------
// Generic WMMA GEMM: C[M,64] = act(A[M,K] @ W[K,64] + bias)
// One wave computes a 16x16 tile with V_WMMA_F32_16X16X4_F32, accumulating
// over K in steps of 4.  Block = 128 threads = 4 waves = one 16x64 row stripe.
//
// ISA fragment layouts (CDNA5 ISA):
//   A 16x4 f32 : lane m=lane&15, VGPR0=A[m][2*half], VGPR1=A[m][2*half+1]
//   B 4x16 f32 : lane n=lane&15, VGPR0=B[2*half][n], VGPR1=B[2*half+1][n]
//   C/D 16x16  : VGPR j = C[j + 8*half][n]
// ---------------------------------------------------------------------------
template <int K, bool RELU>
__global__ __launch_bounds__(128) void gemm_wmma_kernel(
    const float* __restrict__ A, const float* __restrict__ W,
    const float* __restrict__ bias, float* __restrict__ C, int M) {
  const int lane = threadIdx.x & 31;
  const int wv   = threadIdx.x >> 5;  // column tile 0..3
  const int m    = lane & 15;         // A-row coord == B/C col coord
  const int half = lane >> 4;         // 0 or 1
  const int row0 = blockIdx.x << 4;
  const int n0   = wv << 4;
  if (row0 >= M) return;  // M is a multiple of 16 here (100000)

  v8f acc;
  const float b = bias[n0 + m];
#pragma unroll
  for (int j = 0; j < 8; ++j) acc[j] = b;

  const float* ap = A + (size_t)(row0 + m) * K + 2 * half;
  const float* wp = W + (size_t)(2 * half) * HDIM + n0 + m;
#pragma unroll
  for (int k = 0; k < K; k += 4) {
    const float2 av = *(const float2*)(ap + k);
    v2f afrag;
    afrag.x = av.x;
    afrag.y = av.y;
    v2f bfrag;
    bfrag.x = wp[(size_t)k * HDIM];
    bfrag.y = wp[(size_t)(k + 1) * HDIM];
    acc = __builtin_amdgcn_wmma_f32_16x16x4_f32(false, afrag, false, bfrag,
                                                (short)0, acc, false, false);
  }

  float* cp = C + (size_t)(row0 + 8 * half) * HDIM + n0 + m;
#pragma unroll
  for (int j = 0; j < 8; ++j) {
    float v = acc[j];
    if (RELU) v = fmaxf(v, 0.0f);
    cp[(size_t)j * HDIM] = v;
  }
}

// ---------------------------------------------------------------------------
// Fused edge kernel: one wave per edge (lane owns channels 2*lane, 2*lane+1).
//   e_c  = ep_b + edge_attr[e] @ ep_w      (recomputed, 16x64 MACs/edge)
//   msg  = relu(h[src] + e_c)
//   zbuf[dst] += msg                        (zbuf preloaded with h)
// edge_attr loads are wave-uniform (scalarizable); ep_w staged in LDS.
// ---------------------------------------------------------------------------
__global__ __launch_bounds__(256) void edge_msg_kernel(
    const float* __restrict__ h, const float* __restrict__ edge_attr,
    const int* __restrict__ eidx, const float* __restrict__ ep_w,
    const float* __restrict__ ep_b, float* __restrict__ zbuf, int E) {
  __shared__ float2 lw[16 * 32];
  for (int i = threadIdx.x; i < 512; i += blockDim.x)
    lw[i] = ((const float2*)ep_w)[i];
  __syncthreads();

  const int lane    = threadIdx.x & 31;
  const float2 eb   = ((const float2*)ep_b)[lane];
  const int wave    = (blockIdx.x * blockDim.x + threadIdx.x) >> 5;
  const int nwaves  = (gridDim.x * blockDim.x) >> 5;

  for (int e = wave; e < E; e += nwaves) {
    const int src = eidx[e];
    const int dst = eidx[E + e];
    const float* ea = edge_attr + (size_t)e * 16;
    if (e + nwaves < E)  // emits global_prefetch_b8 for next edge's attrs
      __builtin_prefetch(edge_attr + (size_t)(e + nwaves) * 16, 0, 1);

    float2 acc = eb;
#pragma unroll
    for (int k = 0; k < 16; ++k) {
      const float a  = ea[k];  // wave-uniform
      const float2 w = lw[k * 32 + lane];
      acc.x = fmaf(a, w.x, acc.x);
      acc.y = fmaf(a, w.y, acc.y);
    }
    const float2 hv = *(const float2*)(h + (size_t)src * HDIM + 2 * lane);
    const float mx = fmaxf(hv.x + acc.x, 0.0f);
    const float my = fmaxf(hv.y + acc.y, 0.0f);
    float* zp = zbuf + (size_t)dst * HDIM + 2 * lane;
    atomicAdd(zp, mx);
    atomicAdd(zp + 1, my);
  }
}

// ---------------------------------------------------------------------------
// BatchNorm over the node axis: partial sums/sumsq per block via LDS, then
// 128 atomics/block into stats[0:64]=sum, stats[64:128]=sumsq.
// ---------------------------------------------------------------------------
__global__ __launch_bounds__(256) void bn_stats_kernel(
    const float* __restrict__ z, float* __restrict__ stats, int M) {
  const int t  = threadIdx.x;
  const int c  = t & 63;
  const int r0 = t >> 6;  // 0..3
  float s = 0.0f, q = 0.0f;
  for (int row = blockIdx.x * 4 + r0; row < M; row += gridDim.x * 4) {
    const float v = z[(size_t)row * HDIM + c];
    s += v;
    q = fmaf(v, v, q);
  }
  __shared__ float ls[256], lq[256];
  ls[t] = s;
  lq[t] = q;
  __syncthreads();
  if (t < 64) {
    s = ls[t] + ls[t + 64] + ls[t + 128] + ls[t + 192];
    q = lq[t] + lq[t + 64] + lq[t + 128] + lq[t + 192];
    atomicAdd(&stats[c], s);
    atomicAdd(&stats[64 + c], q);
  }
}

__global__ void bn_finalize_kernel(const float* __restrict__ stats,
                                   const float* __restrict__ g,
                                   const float* __restrict__ b,
                                   float* __restrict__ coef, int M) {
  const int c = threadIdx.x;  // 64 threads
  const float inv = 1.0f / (float)M;
  const float mu  = stats[c] * inv;
  float var = stats[64 + c] * inv - mu * mu;
  var = fmaxf(var, 0.0f);
  const float sc = g[c] * rsqrtf(var + 1e-5f);
  coef[c]      = sc;
  coef[64 + c] = fmaf(-mu, sc, b[c]);
}

__global__ __launch_bounds__(256) void bn_apply_relu_kernel(
    const float* __restrict__ z, const float* __restrict__ coef,
    float* __restrict__ hout, int total) {
  const int c   = threadIdx.x & 63;  // i%64 == t%64 (strides are mult. of 64)
  const float sc = coef[c];
  const float sh = coef[64 + c];
  for (int i = blockIdx.x * blockDim.x + threadIdx.x; i < total;
       i += gridDim.x * blockDim.x)
    hout[i] = fmaxf(fmaf(z[i], sc, sh), 0.0f);
}

// ---------------------------------------------------------------------------
// Pooling
// ---------------------------------------------------------------------------
__global__ void zero_kernel(float* __restrict__ p, int n) {
  const int i = blockIdx.x * blockDim.x + threadIdx.x;
  if (i < n) p[i] = 0.0f;
}

__global__ __launch_bounds__(256) void pool_accum_kernel(
    const float* __restrict__ h, const int* __restrict__ batch,
    float* __restrict__ psum, float* __restrict__ pcnt, int N) {
  const int lane   = threadIdx.x & 31;
  const int wave   = (blockIdx.x * blockDim.x + threadIdx.x) >> 5;
  const int nwaves = (gridDim.x * blockDim.x) >> 5;
  for (int row = wave; row < N; row += nwaves) {
    const int g = batch[row];
    const float2 hv = *(const float2*)(h + (size_t)row * HDIM + 2 * lane);
    float* sp = psum + (size_t)g * HDIM + 2 * lane;
    atomicAdd(sp, hv.x);
    atomicAdd(sp + 1, hv.y);
    if (lane == 0) atomicAdd(&pcnt[g], 1.0f);
  }
}

__global__ void pool_final_kernel(const float* __restrict__ psum,
                                  const float* __restrict__ pcnt,
                                  float* __restrict__ out) {
  const int i = blockIdx.x * blockDim.x + threadIdx.x;
  if (i < GDIM * HDIM) out[i] = psum[i] / fmaxf(pcnt[i >> 6], 1.0f);
}

// ---------------------------------------------------------------------------
extern "C" void kernel_launch(void* const* d_in, const int* in_sizes, int n_in,
                              void* d_out, int out_size, void* d_ws,
                              size_t ws_size, hipStream_t stream) {
  const float* x         = (const float*)d_in[0];
  const float* edge_attr = (const float*)d_in[1];
  const int*   eidx      = (const int*)d_in[2];
  const int*   batch     = (const int*)d_in[3];
  const float* np_w      = (const float*)d_in[4];
  const float* np_b      = (const float*)d_in[5];
  const float* ep_w      = (const float*)d_in[6];
  const float* ep_b      = (const float*)d_in[7];
  const float* mlp_w1    = (const float*)d_in[8];
  const float* mlp_b1    = (const float*)d_in[9];
  const float* mlp_w2    = (const float*)d_in[10];
  const float* mlp_b2    = (const float*)d_in[11];
  const float* bn_g      = (const float*)d_in[12];
  const float* bn_b      = (const float*)d_in[13];

  const int N = in_sizes[0] / 128;  // 100000
  const int E = in_sizes[1] / 16;   // 3200000

  float* hbuf = (float*)d_out;               // final h lives in d_out
  float* gout = hbuf + (size_t)N * HDIM;     // graph_emb [G,64]

  float* zbuf  = (float*)d_ws;               // [N,64]
  float* tbuf  = zbuf + (size_t)N * HDIM;    // [N,64]
  float* stats = tbuf + (size_t)N * HDIM;    // 128
  float* coef  = stats + 128;                // 128
  float* psum  = coef + 128;                 // G*64
  float* pcnt  = psum + GDIM * HDIM;         // G  (contiguous with psum)

  const int mtiles = (N + 15) / 16;

  // h = x @ np_w + np_b
  gemm_wmma_kernel<128, false><<<mtiles, 128, 0, stream>>>(x, np_w, np_b, hbuf, N);

  for (int l = 0; l < 3; ++l) {
    // z := h, then scatter-add relu(h[src] + e) into it  =>  z = h + agg
    hipMemcpyAsync(zbuf, hbuf, (size_t)N * HDIM * sizeof(float),
                   hipMemcpyDeviceToDevice, stream);
    edge_msg_kernel<<<1024, 256, 0, stream>>>(hbuf, edge_attr, eidx, ep_w,
                                              ep_b, zbuf, E);
    // MLP
    gemm_wmma_kernel<64, true><<<mtiles, 128, 0, stream>>>(
        zbuf, mlp_w1 + l * HDIM * HDIM, mlp_b1 + l * HDIM, tbuf, N);
    gemm_wmma_kernel<64, false><<<mtiles, 128, 0, stream>>>(
        tbuf, mlp_w2 + l * HDIM * HDIM, mlp_b2 + l * HDIM, zbuf, N);
    // BatchNorm + ReLU -> h
    zero_kernel<<<1, 128, 0, stream>>>(stats, 128);
    bn_stats_kernel<<<1024, 256, 0, stream>>>(zbuf, stats, N);
    bn_finalize_kernel<<<1, 64, 0, stream>>>(stats, bn_g + l * HDIM,
                                             bn_b + l * HDIM, coef, N);
    bn_apply_relu_kernel<<<2048, 256, 0, stream>>>(zbuf, coef, hbuf, N * HDIM);
  }

  // global mean pool
  zero_kernel<<<(GDIM * HDIM + GDIM + 255) / 256, 256, 0, stream>>>(
      psum, GDIM * HDIM + GDIM);
  pool_accum_kernel<<<512, 256, 0, stream>>>(hbuf, batch, psum, pcnt, N);
  pool_final_kernel<<<(GDIM * HDIM + 255) / 256, 256, 0, stream>>>(psum, pcnt,
                                                                   gout);
}